// CustomModel_7378753814828
// MI455X (gfx1250) — compile-verified
//
#include <hip/hip_runtime.h>
#include <math.h>

typedef float v2f __attribute__((ext_vector_type(2)));
typedef float v8f __attribute__((ext_vector_type(8)));

#define RDIM 4096
#define FDIM 256
#define NK 8
#define KCLAMP_MIN 1e-6f
#define KCLAMP_MAX 1e6f

// ---------------------------------------------------------------------------
// WMMA f32 16x16x4: one wave computes a 16x64 tile of G = x1 @ x2^T.
// A layout (16x4 f32): lane holds A[m = lane&15, k = 2*(lane>>4)+v], v=0,1
// B layout (4x16 f32): lane holds B[k = 2*(lane>>4)+v, n = lane&15]
// Both map to a float2 load at x[(row)*256 + f0 + 2*(lane>>4)].
// ---------------------------------------------------------------------------
__device__ __forceinline__ void gemm_16x64(const float* __restrict__ x1,
                                           const float* __restrict__ x2,
                                           int i0, int j0, int lane, v8f acc[4]) {
  const int half = lane >> 4;
  const int col  = lane & 15;
  const float* ap = x1 + (size_t)(i0 + col) * FDIM + 2 * half;
  const float* bp = x2 + (size_t)(j0 + col) * FDIM + 2 * half;
#pragma unroll
  for (int t = 0; t < 4; ++t) { v8f z = {}; acc[t] = z; }
#pragma unroll 4
  for (int kk = 0; kk < FDIM / 4; ++kk) {
    v2f a  = *(const v2f*)(ap + 4 * kk);
    v2f b0 = *(const v2f*)(bp + 4 * kk);
    v2f b1 = *(const v2f*)(bp + 16 * FDIM + 4 * kk);
    v2f b2 = *(const v2f*)(bp + 32 * FDIM + 4 * kk);
    v2f b3 = *(const v2f*)(bp + 48 * FDIM + 4 * kk);
    acc[0] = __builtin_amdgcn_wmma_f32_16x16x4_f32(false, a, false, b0, (short)0, acc[0], false, false);
    acc[1] = __builtin_amdgcn_wmma_f32_16x16x4_f32(false, a, false, b1, (short)0, acc[1], false, false);
    acc[2] = __builtin_amdgcn_wmma_f32_16x16x4_f32(false, a, false, b2, (short)0, acc[2], false, false);
    acc[3] = __builtin_amdgcn_wmma_f32_16x16x4_f32(false, a, false, b3, (short)0, acc[3], false, false);
  }
}

// ---------------------------------------------------------------------------
// Row stats: sum and sum-of-squares for every row of x1 (rows 0..4095) and
// x2 (rows 4096..8191). One wave per row, 8 floats per lane.
// ---------------------------------------------------------------------------
__global__ __launch_bounds__(256) void stats_k(const float* __restrict__ x1,
                                               const float* __restrict__ x2,
                                               float* __restrict__ r1, float* __restrict__ s1,
                                               float* __restrict__ r2, float* __restrict__ s2) {
  const int wid  = threadIdx.x >> 5;
  const int lane = threadIdx.x & 31;
  const int row  = blockIdx.x * 8 + wid;
  const float* x = (row < RDIM) ? (x1 + (size_t)row * FDIM)
                                : (x2 + (size_t)(row - RDIM) * FDIM);
  float4 a = *(const float4*)(x + lane * 8);
  float4 b = *(const float4*)(x + lane * 8 + 4);
  float s = a.x + a.y + a.z + a.w + b.x + b.y + b.z + b.w;
  float q = a.x * a.x + a.y * a.y + a.z * a.z + a.w * a.w +
            b.x * b.x + b.y * b.y + b.z * b.z + b.w * b.w;
#pragma unroll
  for (int m = 1; m < 32; m <<= 1) {
    s += __shfl_xor(s, m, 32);
    q += __shfl_xor(q, m, 32);
  }
  if (lane == 0) {
    if (row < RDIM) { r1[row] = q; s1[row] = s; }
    else            { r2[row - RDIM] = q; s2[row - RDIM] = s; }
  }
}

// ---------------------------------------------------------------------------
// Fused GEMM + epilogue. Grid (256, 8): block = 16 rows x 512 cols, 8 waves.
// MODE 0: per-(k,i) chunk-min of clamped dist        -> part
// MODE 1: per-(k,i) chunk-sum of exp(inv*(dmin-dist))-> part
// MODE 2: out[i,j] = sum_k W[k][i]*exp(inv*(dmin-dist))
// ---------------------------------------------------------------------------
template <int MODE>
__global__ __launch_bounds__(256) void pass_k(
    const float* __restrict__ x1, const float* __restrict__ x2,
    const float* __restrict__ r1, const float* __restrict__ s1,
    const float* __restrict__ r2, const float* __restrict__ s2,
    const float* __restrict__ means, const float* __restrict__ sigmas,
    const float* __restrict__ dmin_g, const float* __restrict__ w_g,
    float* __restrict__ part, float* __restrict__ out) {
  const int i0   = blockIdx.x * 16;
  const int c    = blockIdx.y;
  const int wid  = threadIdx.x >> 5;
  const int lane = threadIdx.x & 31;
  const int half = lane >> 4;
  const int col  = lane & 15;
  const int j0   = c * 512 + wid * 64;

  v8f acc[4];
  gemm_16x64(x1, x2, i0, j0, lane, acc);

  float rr[8], sr[8];
#pragma unroll
  for (int v = 0; v < 8; ++v) {
    const int iv = i0 + 8 * half + v;
    rr[v] = r1[iv];
    sr[v] = s1[iv];
  }

  float mk2[NK], mc[NK], inv2s2[NK];
#pragma unroll
  for (int k = 0; k < NK; ++k) {
    const float m = means[k];
    mk2[k] = 2.0f * m;
    mc[k]  = (float)FDIM * m * m;
    if constexpr (MODE >= 1) {
      const float sg = sigmas[k];
      inv2s2[k] = 1.0f / (2.0f * sg * sg);
    }
  }

  float dmv[NK][8];
  float wv[NK][8];
  if constexpr (MODE >= 1) {
#pragma unroll
    for (int k = 0; k < NK; ++k)
#pragma unroll
      for (int v = 0; v < 8; ++v) {
        const int iv = i0 + 8 * half + v;
        dmv[k][v] = dmin_g[(k << 12) + iv];
        if constexpr (MODE == 2) wv[k][v] = w_g[(k << 12) + iv];
      }
  }
  (void)dmv; (void)wv;

  float red[NK][8];
  if constexpr (MODE < 2) {
#pragma unroll
    for (int k = 0; k < NK; ++k)
#pragma unroll
      for (int v = 0; v < 8; ++v) red[k][v] = (MODE == 0) ? 3.0e38f : 0.0f;
  }

#pragma unroll
  for (int t = 0; t < 4; ++t) {
    const int j = j0 + 16 * t + col;
    const float r2j = r2[j];
    const float s2j = s2[j];
#pragma unroll
    for (int v = 0; v < 8; ++v) {
      const float base = rr[v] + r2j - 2.0f * acc[t][v];
      const float sv   = sr[v] - s2j;
      float o = 0.0f;
#pragma unroll
      for (int k = 0; k < NK; ++k) {
        float dist = base - mk2[k] * sv + mc[k];
        dist = fminf(fmaxf(dist, KCLAMP_MIN), KCLAMP_MAX);
        if constexpr (MODE == 0) {
          red[k][v] = fminf(red[k][v], dist);
        } else if constexpr (MODE == 1) {
          const float xx = fminf(inv2s2[k] * (dmv[k][v] - dist), 0.0f);  // NaN-safe
          red[k][v] += __expf(xx);
        } else {
          const float xx = fminf(inv2s2[k] * (dmv[k][v] - dist), 0.0f);
          o += wv[k][v] * __expf(xx);
        }
      }
      if constexpr (MODE == 2) {
        const int iv = i0 + 8 * half + v;
        out[(size_t)iv * RDIM + j] = o;
      }
    }
  }

  if constexpr (MODE < 2) {
    // Reduce over j within the wave: butterfly across each 16-lane half
    // (lanes 0-15 hold rows i0..i0+7, lanes 16-31 hold rows i0+8..i0+15).
#pragma unroll
    for (int k = 0; k < NK; ++k)
#pragma unroll
      for (int v = 0; v < 8; ++v) {
        float x = red[k][v];
#pragma unroll
        for (int m = 1; m <= 8; m <<= 1) {
          const float y = __shfl_xor(x, m, 32);
          x = (MODE == 0) ? fminf(x, y) : (x + y);
        }
        red[k][v] = x;
      }
    __shared__ float lred[8][NK][16];
    if (col == 0) {
#pragma unroll
      for (int k = 0; k < NK; ++k)
#pragma unroll
        for (int v = 0; v < 8; ++v) lred[wid][k][8 * half + v] = red[k][v];
    }
    __syncthreads();
    // Deterministic combine across the 8 waves of this block.
    if (threadIdx.x < 128) {
      const int k   = threadIdx.x >> 4;
      const int row = threadIdx.x & 15;
      float x = lred[0][k][row];
#pragma unroll
      for (int w = 1; w < 8; ++w)
        x = (MODE == 0) ? fminf(x, lred[w][k][row]) : (x + lred[w][k][row]);
      part[((size_t)(c * NK + k) << 12) + i0 + row] = x;
    }
  }
}

// ---------------------------------------------------------------------------
// Combine chunk partials.
// ---------------------------------------------------------------------------
__global__ __launch_bounds__(256) void reduce_min_k(const float* __restrict__ part,
                                                    float* __restrict__ dmin) {
  const int t = blockIdx.x * blockDim.x + threadIdx.x;  // 0 .. 8*4096-1
  if (t >= NK * RDIM) return;
  const int k = t >> 12;
  const int i = t & (RDIM - 1);
  float m = 3.0e38f;
#pragma unroll
  for (int c = 0; c < 8; ++c) m = fminf(m, part[((size_t)(c * NK + k) << 12) + i]);
  dmin[t] = m;
}

__global__ __launch_bounds__(256) void reduce_z_k(const float* __restrict__ part,
                                                  const float* __restrict__ sp,
                                                  float* __restrict__ W) {
  const int t = blockIdx.x * blockDim.x + threadIdx.x;
  if (t >= NK * RDIM) return;
  const int k = t >> 12;
  const int i = t & (RDIM - 1);
  float z = 0.0f;
#pragma unroll
  for (int c = 0; c < 8; ++c) z += part[((size_t)(c * NK + k) << 12) + i];
  // normalized_weights = softmax(1/sp^2), stabilized
  float wgt[NK], wmax = -3.0e38f;
#pragma unroll
  for (int kk = 0; kk < NK; ++kk) {
    const float s = sp[kk];
    wgt[kk] = 1.0f / (s * s);
    wmax = fmaxf(wmax, wgt[kk]);
  }
  float den = 0.0f;
#pragma unroll
  for (int kk = 0; kk < NK; ++kk) den += __expf(wgt[kk] - wmax);
  const float nw = __expf(wgt[k] - wmax) / den;
  W[t] = nw / z;
}

// ---------------------------------------------------------------------------
// Launch: stats -> pass0(min) -> reduce_min -> pass1(sum) -> reduce_z -> pass2
// Workspace (floats): r1|s1|r2|s2 (16384), partials (262144, reused by pass1),
// dmin (32768), W (32768)  => ~1.32 MB.
// ---------------------------------------------------------------------------
extern "C" void kernel_launch(void* const* d_in, const int* in_sizes, int n_in,
                              void* d_out, int out_size, void* d_ws, size_t ws_size,
                              hipStream_t stream) {
  (void)in_sizes; (void)n_in; (void)out_size; (void)ws_size;
  const float* x1     = (const float*)d_in[0];
  const float* x2     = (const float*)d_in[1];
  const float* sigmas = (const float*)d_in[2];
  const float* means  = (const float*)d_in[3];
  const float* sp     = (const float*)d_in[4];
  float* out = (float*)d_out;

  float* ws   = (float*)d_ws;
  float* r1   = ws;
  float* s1   = ws + RDIM;
  float* r2   = ws + 2 * RDIM;
  float* s2   = ws + 3 * RDIM;
  float* part = ws + 4 * RDIM;                 // 8 chunks * 8 k * 4096 = 262144
  float* dmin = part + 8 * NK * RDIM;          // 32768
  float* W    = dmin + NK * RDIM;              // 32768

  stats_k<<<(2 * RDIM) / 8, 256, 0, stream>>>(x1, x2, r1, s1, r2, s2);

  dim3 grid(RDIM / 16, 8);
  pass_k<0><<<grid, 256, 0, stream>>>(x1, x2, r1, s1, r2, s2, means, sigmas,
                                      nullptr, nullptr, part, nullptr);
  reduce_min_k<<<(NK * RDIM) / 256, 256, 0, stream>>>(part, dmin);
  pass_k<1><<<grid, 256, 0, stream>>>(x1, x2, r1, s1, r2, s2, means, sigmas,
                                      dmin, nullptr, part, nullptr);
  reduce_z_k<<<(NK * RDIM) / 256, 256, 0, stream>>>(part, sp, W);
  pass_k<2><<<grid, 256, 0, stream>>>(x1, x2, r1, s1, r2, s2, means, sigmas,
                                      dmin, W, nullptr, out);
}